// IGCN_Link_Pred_Node_51264729645497
// MI455X (gfx1250) — compile-verified
//
#include <hip/hip_runtime.h>
#include <hip/hip_bf16.h>

// ---------------------------------------------------------------------------
// IGCN link prediction for MI455X (gfx1250, wave32).
//  - Dense GEMMs use V_WMMA_F32_16X16X4_F32; weights staged in LDS via
//    GLOBAL_LOAD_ASYNC_TO_LDS_B128 (ASYNCcnt). Weights are pre-interleaved in
//    workspace as K-pairs ({W[k][n],W[k+1][n]} adjacent) with bank padding
//    (PS = 2N+32, PS mod 64 = 32) so each B fragment is ONE aligned
//    ds_load_b64 straight into the even VGPR pair WMMA wants -> no register
//    re-pairing movs, no bank conflicts between half-waves.
//  - SpMM: lane-parallel float4 gather + global_atomic_add_f32 scatter-add
//    (support matrices fit in the 192MB L2, so this is L2-resident traffic).
//  - Bias pre-seeded into agg buffers; ReLU fused into layer-2 WMMA A-load.
//  - Decoder algebraically collapsed (no inter-layer activation) to one
//    128-wide dot per pair.
// ---------------------------------------------------------------------------

typedef float  v2f __attribute__((ext_vector_type(2)));
typedef float  v8f __attribute__((ext_vector_type(8)));
typedef int    gvec4i __attribute__((vector_size(4 * sizeof(int))));  // matches builtin

#define NFEAT 128
#define NH1   128
#define NH2   64

#if defined(__HIP_DEVICE_COMPILE__) && \
    __has_builtin(__builtin_amdgcn_global_load_async_to_lds_b128)
#define HAVE_ASYNC_LDS 1
#else
#define HAVE_ASYNC_LDS 0
#endif

// ---- pre-pass: W[K,N] row-major -> K-pair interleaved, bank-padded image ----
// Wi[(k/2)*PS + 2n + (k&1)] = W[k*N + n],  PS = 2N+32
template <int K, int N>
__global__ void interleave_W(const float* __restrict__ W, float* __restrict__ Wi)
{
    constexpr int PS = 2 * N + 32;
    const int i = (int)(blockIdx.x * blockDim.x + threadIdx.x);
    if (i >= K * N) return;
    const int k = i / N, n = i % N;
    Wi[(k >> 1) * PS + 2 * n + (k & 1)] = W[i];
}

// ---------------- WMMA GEMM: C[M,N] = (relu?)A[M,K] @ W[K,N] ----------------
// One wave computes a 16x64 strip (4 accumulators reuse the A fragment).
// Wi (interleaved weight image) is async-staged into LDS once per block.
template <bool RELU_A, int K, int N>
__global__ void gemm_wmma_f32(const float* __restrict__ A,
                              const float* __restrict__ Wi,  // interleaved image
                              float* __restrict__ C, int M)
{
    constexpr int PS    = 2 * N + 32;        // floats per K-pair row (bank-padded)
    constexpr int TOTAL = (K / 2) * PS;      // floats in the staged image
    __shared__ float sW[TOTAL];

    // ---- stage interleaved W into LDS (async tensor path when available) ----
    for (int i = (int)threadIdx.x * 4; i < TOTAL; i += (int)blockDim.x * 4) {
#if HAVE_ASYNC_LDS
        __builtin_amdgcn_global_load_async_to_lds_b128(
            (__attribute__((address_space(1))) gvec4i*)(Wi + i),
            (__attribute__((address_space(3))) gvec4i*)(sW + i),
            /*offset=*/0, /*cpol=*/0);
#else
        *(float4*)(sW + i) = *(const float4*)(Wi + i);
#endif
    }
#if HAVE_ASYNC_LDS
#if __has_builtin(__builtin_amdgcn_s_wait_asynccnt)
    __builtin_amdgcn_s_wait_asynccnt(0);
#else
    asm volatile("s_wait_asynccnt 0" ::: "memory");
#endif
#endif
    __syncthreads();

    const int wave = (int)((blockIdx.x * blockDim.x + threadIdx.x) >> 5);
    const int lane = (int)(threadIdx.x & 31);

    constexpr int stripsPerRow = N >> 6;             // N / 64
    const int totalWaves = (M >> 4) * stripsPerRow;
    if (wave >= totalWaves) return;                  // wave-uniform: EXEC stays full

    const int tm   = (wave / stripsPerRow) << 4;
    const int tn   = (wave % stripsPerRow) << 6;
    const int half = lane >> 4;                      // 0: lanes 0-15, 1: lanes 16-31
    const int l    = lane & 15;

    // A fragment base: one M-row per lane, K pair selected by half-wave.
    const float* Arow = A + (size_t)(tm + l) * K + 2 * half;

    // B fragment base in LDS: pair-row (k0/2 + half), column tn+l.
    // Half-wave pair-rows differ by PS (mod 64 = 32) -> disjoint bank halves.
    const float* Bbase = sW + half * PS + 2 * (tn + l);

    v8f acc[4] = {{}, {}, {}, {}};

#pragma unroll 8
    for (int k0 = 0; k0 < K; k0 += 4) {
        v2f a = *(const v2f*)(Arow + k0);            // A[row][k0+2h], A[row][k0+2h+1]
        if (RELU_A) { a.x = fmaxf(a.x, 0.0f); a.y = fmaxf(a.y, 0.0f); }

        const float* Bp = Bbase + (k0 >> 1) * PS;    // {W[ka][col], W[ka+1][col]}
#pragma unroll
        for (int t = 0; t < 4; ++t) {
            v2f b = *(const v2f*)(Bp + t * 32);      // one ds_load_b64 per fragment
            acc[t] = __builtin_amdgcn_wmma_f32_16x16x4_f32(
                false, a, false, b, (short)0, acc[t], false, false);
        }
    }

    // C/D layout: VGPR r -> row tm + r + 8*half, col tn + t*16 + l
#pragma unroll
    for (int t = 0; t < 4; ++t) {
#pragma unroll
        for (int r = 0; r < 8; ++r) {
            C[(size_t)(tm + r + 8 * half) * N + tn + t * 16 + l] = acc[t][r];
        }
    }
}

// ---------------- bias broadcast init: dst[i*F + j] = b[j] -----------------
__global__ void fill_bias(float* __restrict__ dst, const float* __restrict__ b,
                          int Fmask, size_t total)
{
    size_t i = (size_t)blockIdx.x * blockDim.x + threadIdx.x;
    const size_t stride = (size_t)gridDim.x * blockDim.x;
    for (; i < total; i += stride)
        dst[i] = b[i & (size_t)Fmask];
}

// ---------------- SpMM scatter-add: agg[dst] += val * support[src] ----------
// FEATS/4 lanes per edge, float4 gather, hardware f32 atomics into L2.
template <int FEATS>
__global__ void spmm_atomic(const float* __restrict__ support,
                            const int* __restrict__ edges,   // [2,E]: dst row then src row
                            const float* __restrict__ vals,
                            float* __restrict__ agg, int E)
{
    constexpr int TPE = FEATS / 4;
    const int e = (int)(blockIdx.x * (blockDim.x / TPE) + threadIdx.x / TPE);
    const int t = (int)(threadIdx.x % TPE);
    if (e >= E) return;

    const int   dst = edges[e];
    const int   src = edges[E + e];
    const float v   = vals[e];

    const float4 s = *(const float4*)(support + (size_t)src * FEATS + t * 4);
    float* out = agg + (size_t)dst * FEATS + t * 4;
    unsafeAtomicAdd(out + 0, s.x * v);
    unsafeAtomicAdd(out + 1, s.y * v);
    unsafeAtomicAdd(out + 2, s.z * v);
    unsafeAtomicAdd(out + 3, s.w * v);
}

// ---------------- gating: g = (x1.ag1) x1 + (x2.ag2) x2 (per node) ---------
__global__ void gating(const float* __restrict__ x1, const float* __restrict__ x2,
                       const float* __restrict__ ag1, const float* __restrict__ ag2,
                       float* __restrict__ g, int Nn)
{
    const int wave = (int)((blockIdx.x * blockDim.x + threadIdx.x) >> 5);
    const int lane = (int)(threadIdx.x & 31);
    if (wave >= Nn) return;

    const float* r1 = x1 + (size_t)wave * NH2;
    const float* r2 = x2 + (size_t)wave * NH2;
    const float a0 = r1[lane], a1 = r1[lane + 32];
    const float b0 = r2[lane], b1 = r2[lane + 32];

    float d1 = a0 * ag1[lane] + a1 * ag1[lane + 32];
    float d2 = b0 * ag2[lane] + b1 * ag2[lane + 32];
#pragma unroll
    for (int off = 16; off > 0; off >>= 1) {
        d1 += __shfl_xor(d1, off, 32);
        d2 += __shfl_xor(d2, off, 32);
    }
    g[(size_t)wave * NH2 + lane]      = d1 * a0 + d2 * b0;
    g[(size_t)wave * NH2 + lane + 32] = d1 * a1 + d2 * b1;
}

// ---------------- collapsed decoder weights: weff[128], weff[128]=b_eff ----
__global__ void build_weff(const float* __restrict__ d1W,  // [128,64]
                           const float* __restrict__ d1b,  // [64]
                           const float* __restrict__ d2W,  // [64]
                           const float* __restrict__ d2b,  // [1]
                           float* __restrict__ weff)
{
    const int k = (int)threadIdx.x;                 // 128 threads
    float s = 0.0f;
    for (int j = 0; j < 64; ++j) s += d1W[k * 64 + j] * d2W[j];
    weff[k] = s;
    if (k == 0) {
        float be = d2b[0];
        for (int j = 0; j < 64; ++j) be += d1b[j] * d2W[j];
        weff[128] = be;
    }
}

// ---------------- decoder: out[p] = [g[i0],g[i1]] . weff + b_eff -----------
__global__ void decoder(const float* __restrict__ g, const int* __restrict__ idx,
                        const float* __restrict__ weff, float* __restrict__ out, int Pn)
{
    const int wave = (int)((blockIdx.x * blockDim.x + threadIdx.x) >> 5);
    const int lane = (int)(threadIdx.x & 31);
    if (wave >= Pn) return;

    const int i0 = idx[wave];
    const int i1 = idx[Pn + wave];
    const float* g0 = g + (size_t)i0 * NH2;
    const float* g1 = g + (size_t)i1 * NH2;

    float s = g0[lane] * weff[lane] + g0[lane + 32] * weff[lane + 32]
            + g1[lane] * weff[64 + lane] + g1[lane + 32] * weff[96 + lane];
#pragma unroll
    for (int off = 16; off > 0; off >>= 1) s += __shfl_xor(s, off, 32);
    if (lane == 0) out[wave] = s + weff[128];
}

// ---------------------------------------------------------------------------
extern "C" void kernel_launch(void* const* d_in, const int* in_sizes, int n_in,
                              void* d_out, int out_size, void* d_ws, size_t ws_size,
                              hipStream_t stream)
{
    const float* x       = (const float*)d_in[0];
    const int*   o_edges = (const int*)  d_in[1];
    const float* o_vals  = (const float*)d_in[2];
    const int*   s_edges = (const int*)  d_in[3];
    const float* s_vals  = (const float*)d_in[4];
    const int*   idx     = (const int*)  d_in[5];
    const float* W_o1    = (const float*)d_in[6];
    const float* b_o1    = (const float*)d_in[7];
    const float* W_o2    = (const float*)d_in[8];
    const float* b_o2    = (const float*)d_in[9];
    const float* W_s1    = (const float*)d_in[10];
    const float* b_s1    = (const float*)d_in[11];
    const float* W_s2    = (const float*)d_in[12];
    const float* b_s2    = (const float*)d_in[13];
    const float* ag1     = (const float*)d_in[14];
    const float* ag2     = (const float*)d_in[15];
    const float* dec1_W  = (const float*)d_in[16];
    const float* dec1_b  = (const float*)d_in[17];
    const float* dec2_W  = (const float*)d_in[18];
    const float* dec2_b  = (const float*)d_in[19];
    float* out = (float*)d_out;

    const int N = in_sizes[0] / NFEAT;   // 100000
    const int E = in_sizes[2];           // 1600000
    const int P = in_sizes[5] / 2;       // 100000

    // -------- workspace layout (floats) --------
    float* S1   = (float*)d_ws;          // [N,128] support (both layer-1 GEMMs)
    float* AG1  = S1  + (size_t)N * NH1; // [N,128] layer-1 aggregate
    float* S2   = AG1 + (size_t)N * NH1; // [N,64]  layer-2 support (reused as g)
    float* X1   = S2  + (size_t)N * NH2; // [N,64]  branch-1 output
    float* X2   = X1  + (size_t)N * NH2; // [N,64]  branch-2 output
    float* WEFF = X2  + (size_t)N * NH2; // [129]   collapsed decoder
    float* WINT = WEFF + 132;            // [<=18432] interleaved weight image (16B aligned)
    float* G    = S2;                    // gating output overlays freed S2

    // -------- launch geometry --------
    const int tilesL1 = (N >> 4) * (NH1 >> 6);       // 16x64 strips, N=128 -> 2/row
    const int tilesL2 = (N >> 4) * (NH2 >> 6);       // N=64 -> 1/row
    const int gemmBlk = 256;                          // 8 waves
    const int gridL1  = (tilesL1 + 7) / 8;
    const int gridL2  = (tilesL2 + 7) / 8;

    const int ilvGrid1 = (NFEAT * NH1 + 255) / 256;  // interleave pre-pass grids
    const int ilvGrid2 = (NH1 * NH2 + 255) / 256;

    const int spmmBlk = 256;
    const int grid128 = (E + (spmmBlk / 32) - 1) / (spmmBlk / 32);  // 32 lanes/edge
    const int grid64  = (E + (spmmBlk / 16) - 1) / (spmmBlk / 16);  // 16 lanes/edge

    const size_t tot128 = (size_t)N * NH1;
    const size_t tot64  = (size_t)N * NH2;
    const int fillGrid  = 2048;

    const int nodeWaveGrid = (N + 7) / 8;   // 256-thread blocks, 8 waves, 1 node/wave
    const int pairWaveGrid = (P + 7) / 8;

    // ================= branch 1: original graph =================
    interleave_W<NFEAT, NH1><<<ilvGrid1, 256, 0, stream>>>(W_o1, WINT);
    gemm_wmma_f32<false, NFEAT, NH1><<<gridL1, gemmBlk, 0, stream>>>(x, WINT, S1, N);
    fill_bias<<<fillGrid, 256, 0, stream>>>(AG1, b_o1, NH1 - 1, tot128);
    spmm_atomic<NH1><<<grid128, spmmBlk, 0, stream>>>(S1, o_edges, o_vals, AG1, E);
    interleave_W<NH1, NH2><<<ilvGrid2, 256, 0, stream>>>(W_o2, WINT);
    gemm_wmma_f32<true, NH1, NH2><<<gridL2, gemmBlk, 0, stream>>>(AG1, WINT, S2, N);
    fill_bias<<<fillGrid, 256, 0, stream>>>(X1, b_o2, NH2 - 1, tot64);
    spmm_atomic<NH2><<<grid64, spmmBlk, 0, stream>>>(S2, o_edges, o_vals, X1, E);

    // ================= branch 2: skip graph =====================
    interleave_W<NFEAT, NH1><<<ilvGrid1, 256, 0, stream>>>(W_s1, WINT);
    gemm_wmma_f32<false, NFEAT, NH1><<<gridL1, gemmBlk, 0, stream>>>(x, WINT, S1, N);
    fill_bias<<<fillGrid, 256, 0, stream>>>(AG1, b_s1, NH1 - 1, tot128);
    spmm_atomic<NH1><<<grid128, spmmBlk, 0, stream>>>(S1, s_edges, s_vals, AG1, E);
    interleave_W<NH1, NH2><<<ilvGrid2, 256, 0, stream>>>(W_s2, WINT);
    gemm_wmma_f32<true, NH1, NH2><<<gridL2, gemmBlk, 0, stream>>>(AG1, WINT, S2, N);
    fill_bias<<<fillGrid, 256, 0, stream>>>(X2, b_s2, NH2 - 1, tot64);
    spmm_atomic<NH2><<<grid64, spmmBlk, 0, stream>>>(S2, s_edges, s_vals, X2, E);

    // ================= gating + decoder =========================
    gating<<<nodeWaveGrid, 256, 0, stream>>>(X1, X2, ag1, ag2, G, N);
    build_weff<<<1, 128, 0, stream>>>(dec1_W, dec1_b, dec2_W, dec2_b, WEFF);
    decoder<<<pairWaveGrid, 256, 0, stream>>>(G, idx, WEFF, out, P);
}